// GATNet_25623774888616
// MI455X (gfx1250) — compile-verified
//
#include <hip/hip_runtime.h>
#include <cstdint>
#include <cstddef>

typedef float v2f __attribute__((ext_vector_type(2)));
typedef float v8f __attribute__((ext_vector_type(8)));

#define LEAKY 0.2f

// ---------------------------------------------------------------------------
// float atomic max via signed/unsigned integer ordering trick
// ---------------------------------------------------------------------------
__device__ __forceinline__ void atomicMaxF(float* addr, float val) {
  if (val >= 0.0f) {
    atomicMax((int*)addr, __float_as_int(val));
  } else {
    atomicMin((unsigned int*)addr, __float_as_uint(val));
  }
}

// ---------------------------------------------------------------------------
// fp32 WMMA GEMM: C[M, NT*16] = A[M,K] * B[K, NT*16]
// - one wave computes a 16-row x (NT*16)-col tile with V_WMMA_F32_16X16X4_F32
// - B panels are staged 64 K-rows at a time into LDS with the CDNA5 async
//   copy path (global_load_async_to_lds_b128 + s_wait_asynccnt), then WMMA
//   fragments are fetched from LDS (ds loads) instead of serialized VMEM.
// A fragment (16x4 fp32): lanes 0-15 -> K=0,1 ; lanes 16-31 -> K=2,3
// B fragment (4x16 fp32): lanes 0-15 -> rows 0,1 ; lanes 16-31 -> rows 2,3
// C/D (16x16 fp32): VGPR j -> M=j (lanes 0-15) / M=j+8 (lanes 16-31), N=lane&15
// ---------------------------------------------------------------------------
template <int NT>
__global__ __launch_bounds__(256) void gemm_wmma(const float* __restrict__ A,
                                                 const float* __restrict__ B,
                                                 float* __restrict__ C,
                                                 int M, int K) {
  constexpr int NC = NT * 16;
  __shared__ float Bt[64 * NC];           // 16 KB (NT=4) / 4 KB (NT=1)

  const int tid   = threadIdx.x;
  const int wave  = blockIdx.x * 8 + (tid >> 5);
  const int lane  = tid & 31;
  const int row0  = wave << 4;
  const bool active = row0 < M;
  const int laneH = lane & 15;            // M index (A) / N index (B,C)
  const int klo   = (lane >> 4) << 1;     // 0 or 2

  v8f acc[NT] = {};

  const int arowIdx = active ? (row0 + laneH) : 0;
  const float* arow = A + (size_t)arowIdx * K + klo;
  const unsigned ldsBase = (unsigned)(size_t)(&Bt[0]);

  for (int k0 = 0; k0 < K; k0 += 64) {
    // --- async-stage 64 x NC panel of B into LDS (128b per lane per shot) ---
    const char* gsrc = (const char*)(B + (size_t)k0 * NC);
#pragma unroll
    for (int j = 0; j < NT; ++j) {
      const int byteOff = (tid + j * 256) * 16;   // float4 granularity
      const unsigned ldsAddr = ldsBase + (unsigned)byteOff;
      const char* gAddr = gsrc + byteOff;
      asm volatile("global_load_async_to_lds_b128 %0, %1, off"
                   :: "v"(ldsAddr), "v"(gAddr)
                   : "memory");
    }
    asm volatile("s_wait_asynccnt 0x0" ::: "memory");
    __syncthreads();

    if (active) {
      for (int kk = 0; kk < 64; kk += 4) {
        v2f a;
        a.x = arow[k0 + kk];
        a.y = arow[k0 + kk + 1];
        const float* brow = &Bt[(kk + klo) * NC + laneH];
#pragma unroll
        for (int t = 0; t < NT; ++t) {
          v2f b;
          b.x = brow[t * 16];
          b.y = brow[NC + t * 16];
          acc[t] = __builtin_amdgcn_wmma_f32_16x16x4_f32(
              /*neg_a=*/false, a, /*neg_b=*/false, b,
              /*c_mod=*/(short)0, acc[t], /*reuse_a=*/false, /*reuse_b=*/false);
        }
      }
    }
    __syncthreads();
  }

  if (active) {
    const int rOff = (lane >> 4) << 3;    // 0 or 8
#pragma unroll
    for (int t = 0; t < NT; ++t) {
#pragma unroll
      for (int j = 0; j < 8; ++j) {
        C[(size_t)(row0 + rOff + j) * NC + t * 16 + laneH] = acc[t][j];
      }
    }
  }
}

// ---------------------------------------------------------------------------
// per-(node,head) attention projections: fs = h.a_self, fn = h.a_neigh
// h laid out [n][HEADS*CH]
// ---------------------------------------------------------------------------
template <int HEADS, int CH>
__global__ void proj_kernel(const float* __restrict__ h,
                            const float* __restrict__ a_self,
                            const float* __restrict__ a_neigh,
                            float* __restrict__ fs, float* __restrict__ fn,
                            int n) {
  long long i = (long long)blockIdx.x * blockDim.x + threadIdx.x;
  long long total = (long long)n * HEADS;
  if (i >= total) return;
  const int hd = (int)(i % HEADS);
  const float* hp = h + i * CH;           // (node*HEADS + hd)*CH
  float s = 0.f, t = 0.f;
#pragma unroll
  for (int c = 0; c < CH; ++c) {
    float v = hp[c];
    s += v * a_self[hd * CH + c];
    t += v * a_neigh[hd * CH + c];
  }
  fs[i] = s;
  fn[i] = t;
}

// ---------------------------------------------------------------------------
// fill
// ---------------------------------------------------------------------------
__global__ void fill_kernel(float* __restrict__ p, float v, long long n) {
  long long i = (long long)blockIdx.x * blockDim.x + threadIdx.x;
  if (i < n) p[i] = v;
}

// ---------------------------------------------------------------------------
// edge pass 1: segment max of leaky_relu(fs[dst]+fn[src]) into m[dst]
// threads over (edge, head); edges >= E are self loops (src=dst=e-E)
// ---------------------------------------------------------------------------
template <int HEADS>
__global__ void edge_max_kernel(const int* __restrict__ src,
                                const int* __restrict__ dst,
                                const float* __restrict__ fs,
                                const float* __restrict__ fn,
                                float* __restrict__ m,
                                long long E, int n) {
  long long i = (long long)blockIdx.x * blockDim.x + threadIdx.x;
  long long total = (E + n) * HEADS;
  if (i >= total) return;
  const int h = (int)(i % HEADS);
  const long long e = i / HEADS;
  int s, d;
  if (e < E) { s = src[e]; d = dst[e]; } else { s = d = (int)(e - E); }
  float z = fs[(long long)d * HEADS + h] + fn[(long long)s * HEADS + h];
  z = z > 0.f ? z : LEAKY * z;
  atomicMaxF(&m[(long long)d * HEADS + h], z);
}

// ---------------------------------------------------------------------------
// edge pass 2: w = exp(e - m[dst]); denom[dst] += w; acc[dst] += w * h[src]
// (softmax normalization folded out of the scatter: out = acc/denom later)
// ---------------------------------------------------------------------------
template <int HEADS, int CH>
__global__ void edge_acc_kernel(const int* __restrict__ src,
                                const int* __restrict__ dst,
                                const float* __restrict__ fs,
                                const float* __restrict__ fn,
                                const float* __restrict__ m,
                                const float* __restrict__ hsrc,
                                float* __restrict__ denom,
                                float* __restrict__ acc,
                                long long E, int n) {
  long long i = (long long)blockIdx.x * blockDim.x + threadIdx.x;
  long long total = (E + n) * HEADS;
  if (i >= total) return;
  const int h = (int)(i % HEADS);
  const long long e = i / HEADS;
  int s, d;
  if (e < E) { s = src[e]; d = dst[e]; } else { s = d = (int)(e - E); }
  float z = fs[(long long)d * HEADS + h] + fn[(long long)s * HEADS + h];
  z = z > 0.f ? z : LEAKY * z;
  const float w = expf(z - m[(long long)d * HEADS + h]);
  atomicAdd(&denom[(long long)d * HEADS + h], w);
  const float* hp = hsrc + ((long long)s * HEADS + h) * CH;
  float* ap = acc + ((long long)d * HEADS + h) * CH;
  float v[CH];
#pragma unroll
  for (int c = 0; c < CH; ++c) v[c] = hp[c];   // vectorizes to b128 loads
#pragma unroll
  for (int c = 0; c < CH; ++c) atomicAdd(&ap[c], w * v[c]);
}

// ---------------------------------------------------------------------------
// finalize layer 1 (in place on acc): elu(acc/denom + b1)
// ---------------------------------------------------------------------------
template <int HEADS, int CH>
__global__ void finalize1_kernel(float* __restrict__ acc,
                                 const float* __restrict__ denom,
                                 const float* __restrict__ b,
                                 int n) {
  long long i = (long long)blockIdx.x * blockDim.x + threadIdx.x;
  long long total = (long long)n * HEADS * CH;
  if (i >= total) return;
  const int hc = (int)(i % (HEADS * CH));
  const long long node = i / (HEADS * CH);
  const int h = hc / CH;
  float v = acc[i] / (denom[node * HEADS + h] + 1e-9f) + b[hc];
  acc[i] = v > 0.f ? v : (expf(v) - 1.f);
}

// ---------------------------------------------------------------------------
// final: out[node] = softmax_16(acc2/denom2 + b2)
// ---------------------------------------------------------------------------
__global__ void final_softmax_kernel(const float* __restrict__ acc2,
                                     const float* __restrict__ denom2,
                                     const float* __restrict__ b2,
                                     float* __restrict__ out, int n) {
  int node = blockIdx.x * blockDim.x + threadIdx.x;
  if (node >= n) return;
  float v[16];
  const float dd = denom2[node] + 1e-9f;
  float mx = -3.402823466e38f;
#pragma unroll
  for (int c = 0; c < 16; ++c) {
    float t = acc2[(long long)node * 16 + c] / dd + b2[c];
    v[c] = t;
    mx = fmaxf(mx, t);
  }
  float s = 0.f;
#pragma unroll
  for (int c = 0; c < 16; ++c) { v[c] = expf(v[c] - mx); s += v[c]; }
  const float inv = 1.f / s;
#pragma unroll
  for (int c = 0; c < 16; ++c) out[(long long)node * 16 + c] = v[c] * inv;
}

// ---------------------------------------------------------------------------
// launch
// ---------------------------------------------------------------------------
extern "C" void kernel_launch(void* const* d_in, const int* in_sizes, int n_in,
                              void* d_out, int out_size, void* d_ws, size_t ws_size,
                              hipStream_t stream) {
  const float* x   = (const float*)d_in[0];
  const float* W1  = (const float*)d_in[1];
  const float* as1 = (const float*)d_in[2];
  const float* an1 = (const float*)d_in[3];
  const float* b1  = (const float*)d_in[4];
  const float* W2  = (const float*)d_in[5];
  const float* as2 = (const float*)d_in[6];
  const float* an2 = (const float*)d_in[7];
  const float* b2  = (const float*)d_in[8];
  const int* esrc  = (const int*)d_in[9];
  const int* edst  = (const int*)d_in[10];

  const int n = in_sizes[0] / 512;       // 100000
  const long long E = in_sizes[9];       // 3.2M
  const int F = 512, HEADS = 8, HC = 64, C2 = 16;

  // workspace layout (floats)
  float* ws = (float*)d_ws;
  size_t o = 0;
  float* h1   = ws + o; o += (size_t)n * HC;
  float* fs1  = ws + o; o += (size_t)n * HEADS;
  float* fn1  = ws + o; o += (size_t)n * HEADS;
  float* m1   = ws + o; o += (size_t)n * HEADS;
  float* dn1  = ws + o; o += (size_t)n * HEADS;
  float* acc1 = ws + o; o += (size_t)n * HC;   // becomes h1p (elu output) in place
  float* h2   = ws + o; o += (size_t)n * C2;
  float* fs2  = ws + o; o += (size_t)n;
  float* fn2  = ws + o; o += (size_t)n;
  float* m2   = ws + o; o += (size_t)n;
  float* dn2  = ws + o; o += (size_t)n;
  float* acc2 = ws + o; o += (size_t)n * C2;

  const int TB = 256;
  const float NEG_INF = -__builtin_inff();
  auto blocks1d = [TB](long long t) { return (unsigned int)((t + TB - 1) / TB); };

  const int waves   = (n + 15) >> 4;       // 16-row tiles
  const int gblocks = (waves + 7) >> 3;    // 8 waves per 256-thread block

  // ---- Layer 1 ----
  gemm_wmma<4><<<gblocks, TB, 0, stream>>>(x, W1, h1, n, F);
  proj_kernel<8, 8><<<blocks1d((long long)n * HEADS), TB, 0, stream>>>(
      h1, as1, an1, fs1, fn1, n);

  fill_kernel<<<blocks1d((long long)n * HEADS), TB, 0, stream>>>(m1, NEG_INF, (long long)n * HEADS);
  fill_kernel<<<blocks1d((long long)n * HEADS), TB, 0, stream>>>(dn1, 0.f, (long long)n * HEADS);
  fill_kernel<<<blocks1d((long long)n * HC), TB, 0, stream>>>(acc1, 0.f, (long long)n * HC);

  const long long et1 = (E + n) * HEADS;
  edge_max_kernel<8><<<blocks1d(et1), TB, 0, stream>>>(esrc, edst, fs1, fn1, m1, E, n);
  edge_acc_kernel<8, 8><<<blocks1d(et1), TB, 0, stream>>>(esrc, edst, fs1, fn1, m1, h1,
                                                          dn1, acc1, E, n);
  finalize1_kernel<8, 8><<<blocks1d((long long)n * HC), TB, 0, stream>>>(acc1, dn1, b1, n);

  // ---- Layer 2 ----
  gemm_wmma<1><<<gblocks, TB, 0, stream>>>(acc1, W2, h2, n, HC);
  proj_kernel<1, 16><<<blocks1d((long long)n), TB, 0, stream>>>(h2, as2, an2, fs2, fn2, n);

  fill_kernel<<<blocks1d((long long)n), TB, 0, stream>>>(m2, NEG_INF, (long long)n);
  fill_kernel<<<blocks1d((long long)n), TB, 0, stream>>>(dn2, 0.f, (long long)n);
  fill_kernel<<<blocks1d((long long)n * C2), TB, 0, stream>>>(acc2, 0.f, (long long)n * C2);

  const long long et2 = (E + n);
  edge_max_kernel<1><<<blocks1d(et2), TB, 0, stream>>>(esrc, edst, fs2, fn2, m2, E, n);
  edge_acc_kernel<1, 16><<<blocks1d(et2), TB, 0, stream>>>(esrc, edst, fs2, fn2, m2, h2,
                                                           dn2, acc2, E, n);
  final_softmax_kernel<<<blocks1d((long long)n), TB, 0, stream>>>(acc2, dn2, b2,
                                                                  (float*)d_out, n);
}